// SIRSDE_63565515981000
// MI455X (gfx1250) — compile-verified
//
#include <hip/hip_runtime.h>
#include <stdint.h>

// ---------------------------------------------------------------------------
// SIR-with-stochastic-R0 SDE + summary stats, MI455X (gfx1250, wave32).
// One thread = one simulation. One wave = 32 contiguous sims.
// Noise streamed via Tensor Data Mover (tensor_load_to_lds), quad-buffered
// in LDS with depth-3 prefetch (19.2 KB in flight per wave ~= 19.6 MB chip-
// wide > BW*latency ~= 12 MB), waited with s_wait_tensorcnt.
// Memory floor ~29 us @ 23.3 TB/s for the 655 MB of consumed noise.
// ---------------------------------------------------------------------------

typedef unsigned int       u32;
typedef unsigned long long u64;
typedef __attribute__((ext_vector_type(4))) u32 u32x4;
typedef __attribute__((ext_vector_type(8))) u32 u32x8;

#define N_SIMS     32768
#define STEPS      50
#define T_CHUNK    50                         // substeps per TDM tile
#define N_CHUNKS   100                        // 100*50 = 5000 substeps (rows 5000..5099 unused by ref)
#define ROW_BYTES  ((u64)N_SIMS * 4ull)       // 131072 B per substep row
#define CHUNK_BYTES ((u64)T_CHUNK * ROW_BYTES)
#define BUF_FLOATS (T_CHUNK * 32)             // 1600 floats = 6400 B per buffer
#define BUF_BYTES  (BUF_FLOATS * 4u)
#define NBUF       4                          // quad buffer, depth-3 prefetch
#define WAVE_FLOATS (NBUF * BUF_FLOATS)       // 6400 floats per wave
#define XS_BASE    (2 * WAVE_FLOATS)          // 2 waves of noise buffers first
#define XS_STRIDE  51                         // 50 + 1 pad -> conflict-free LDS

// Issue one TDM 2D tile load: 32 floats wide x T_CHUNK rows, row stride 32768
// floats, from global 'glo/ghi' into LDS byte address 'lds_addr'.
__device__ __forceinline__ void tdm_load_tile(u32 glo, u32 ghi, u32 lds_addr) {
  u32x4 g0;
  g0[0] = 1u;                                     // count=1, user mode, no gather
  g0[1] = lds_addr;                               // LDS byte address of tile
  g0[2] = glo;                                    // global_addr[31:0]
  g0[3] = (ghi & 0x01FFFFFFu) | 0x80000000u;      // global_addr[56:32] | type=2
  u32x8 g1;
  g1[0] = 0x00020000u;                            // workgroup_mask=0, data_size=2 (4B)
  g1[1] = 0x80000000u;                            // tensor_dim0 = 32768 (low16 @ [31:16])
  g1[2] = (5100u << 16);                          // dim0_hi=0 | tensor_dim1=5100
  g1[3] = (32u << 16);                            // dim1_hi=0 | tile_dim0=32
  g1[4] = (u32)T_CHUNK;                           // tile_dim1=50, tile_dim2=0
  g1[5] = 32768u;                                 // tensor_dim0_stride[31:0]
  g1[6] = 0u;                                     // stride hi | tensor_dim1_stride lo
  g1[7] = 0u;
  asm volatile("tensor_load_to_lds %0, %1" :: "s"(g0), "s"(g1) : "memory");
}

__global__ __launch_bounds__(64)
void sir_sde_summary_kernel(const float* __restrict__ z,
                            const float* __restrict__ noise,
                            const float* __restrict__ u,
                            float* __restrict__ out) {
  __shared__ float smem[XS_BASE + 64 * XS_STRIDE];   // 16064 floats = 64256 B

  const int tid  = threadIdx.x;
  const int lane = tid & 31;
  const u32 wv   = __builtin_amdgcn_readfirstlane((u32)(tid >> 5));
  const int sim  = (int)blockIdx.x * 64 + tid;

  // ---- per-sim parameters (z is [N,4] row-major -> one float4 per sim) ----
  const float4 zz = reinterpret_cast<const float4*>(z)[sim];
  const float infr  = fminf(zz.x, 1.0f);
  const float rec   = fminf(zz.y, 1.0f);
  const float mrev  = fminf(zz.z, 1.0f);
  const float vol   = fminf(zz.w, 1.0f);
  const float r0bar = infr / rec;
  const float recdt  = rec  * 0.01f;       // rec * DT
  const float mrevdt = mrev * 0.01f;       // mrev * DT
  const float volsdt = vol  * 0.1f;        // vol * sqrt(DT)
  const float omm    = 1.0f - mrevdt;      // r0 decay factor
  const float arm    = mrevdt * r0bar;     // reversion constant

  float s = 0.99f, infc = 0.01f, r0 = r0bar;   // 'r' never affects output

  // ---- wave-uniform TDM addressing ----
  const u32 simBase = (u32)blockIdx.x * 64u + wv * 32u;
  const u64 gbase   = (u64)(uintptr_t)noise + (u64)simBase * 4ull;
  const u32 ldsWave = __builtin_amdgcn_readfirstlane(
      (u32)(uintptr_t)&smem[wv * WAVE_FLOATS]);

  // depth-3 prefetch pipeline
  tdm_load_tile((u32)gbase, (u32)(gbase >> 32), ldsWave);
  { const u64 g = gbase + CHUNK_BYTES;
    tdm_load_tile((u32)g, (u32)(g >> 32), ldsWave + BUF_BYTES); }
  { const u64 g = gbase + 2 * CHUNK_BYTES;
    tdm_load_tile((u32)g, (u32)(g >> 32), ldsWave + 2 * BUF_BYTES); }

  float* xsp = &smem[XS_BASE + tid * XS_STRIDE];

  for (int c = 0; c < N_CHUNKS; ++c) {
    if (c + 3 < N_CHUNKS) {
      const u64 g = gbase + (u64)(c + 3) * CHUNK_BYTES;
      tdm_load_tile((u32)g, (u32)(g >> 32),
                    ldsWave + (u32)((c + 3) & 3) * BUF_BYTES);
    }
    if (c < N_CHUNKS - 3)       __builtin_amdgcn_s_wait_tensorcnt(3);
    else if (c == N_CHUNKS - 3) __builtin_amdgcn_s_wait_tensorcnt(2);
    else if (c == N_CHUNKS - 2) __builtin_amdgcn_s_wait_tensorcnt(1);
    else                        __builtin_amdgcn_s_wait_tensorcnt(0);

    const float* nb = &smem[wv * WAVE_FLOATS + (u32)(c & 3) * BUF_FLOATS + lane];
    #pragma unroll
    for (int t = 0; t < T_CHUNK; ++t) {
      const float n   = nb[t * 32];                       // ds_load_b32, no conflicts
      const float t1  = r0 * s;                           // r0*s
      const float sN  = fmaf(-t1, recdt, s);              // s - r0*rec*s*dt
      const float iN  = fmaf(t1 - infc, recdt, infc);     // i + (ni-nr)*dt
      const float r0a = fmaf(r0, omm, arm);               // r0 + mrevdt*(r0bar-r0)
      const float sq  = __builtin_amdgcn_sqrtf(fabsf(r0));
      r0 = fmaf(sq, volsdt * n, r0a);                     // + sqrt(|r0|)*vol*dW
      s = sN; infc = iN;
    }
    if (c & 1)  // every 100 substeps: record infected, pre-clipped & scaled
      xsp[c >> 1] = fmaxf(infc * 10000.0f, 1e-6f);
  }

  // ------------------------------ summary ---------------------------------
  float xv[STEPS];
  #pragma unroll
  for (int k = 0; k < STEPS; ++k) xv[k] = xsp[k];

  float sum = 0.0f, mx = xv[0];
  int am = 0;
  #pragma unroll
  for (int k = 0; k < STEPS; ++k) {
    sum += xv[k];
    if (xv[k] > mx) { mx = xv[k]; am = k; }    // strict > => first argmax
  }
  const float mean = sum / 50.0f;

  // median of 50 = avg of order stats 24 & 25 (rank counting, tie-exact)
  float m24 = 0.0f, m25 = 0.0f;
  for (int j = 0; j < STEPS; ++j) {
    const float v = xsp[j];
    int cl = 0, ce = 0;
    #pragma unroll
    for (int k = 0; k < STEPS; ++k) { cl += (xv[k] < v); ce += (xv[k] == v); }
    const int ch = cl + ce;
    if (cl <= 24 && 24 < ch) m24 = v;
    if (cl <= 25 && 25 < ch) m25 = v;
  }
  const float median = 0.5f * (m24 + m25);

  // lag-1 autocorrelation, clipped at 0
  const float amean = (sum - xv[STEPS - 1]) * (1.0f / 49.0f);
  const float bmean = (sum - xv[0])        * (1.0f / 49.0f);
  float saa = 0.0f, sbb = 0.0f, sab = 0.0f;
  #pragma unroll
  for (int k = 0; k < STEPS - 1; ++k) {
    const float da = xv[k]     - amean;
    const float db = xv[k + 1] - bmean;
    saa = fmaf(da, da, saa);
    sbb = fmaf(db, db, sbb);
    sab = fmaf(da, db, sab);
  }
  float ac = sab / __builtin_amdgcn_sqrtf(saa * sbb);
  ac = fmaxf(ac, 0.0f);

  const float maxat = (float)am + u[sim];

  float* o = out + (size_t)sim * 5;
  o[0] = mean; o[1] = mx; o[2] = median; o[3] = maxat; o[4] = ac;
}

extern "C" void kernel_launch(void* const* d_in, const int* in_sizes, int n_in,
                              void* d_out, int out_size, void* d_ws, size_t ws_size,
                              hipStream_t stream) {
  (void)in_sizes; (void)n_in; (void)out_size; (void)d_ws; (void)ws_size;
  const float* z     = (const float*)d_in[0];
  const float* noise = (const float*)d_in[1];
  const float* u     = (const float*)d_in[2];
  float* out         = (float*)d_out;
  dim3 grid(N_SIMS / 64), block(64);
  hipLaunchKernelGGL(sir_sde_summary_kernel, grid, block, 0, stream,
                     z, noise, u, out);
}